// TemporalFeatureTracker_87205015978029
// MI455X (gfx1250) — compile-verified
//
#include <hip/hip_runtime.h>
#include <math.h>

typedef float v2f __attribute__((ext_vector_type(2)));
typedef float v8f __attribute__((ext_vector_type(8)));

#define STAB_N   3584          // 7*512
#define CAUSAL_OFF 10752       // 3*3584
#define EVO_OFF  76288         // 3*3584 + 65536

__device__ __forceinline__ float sigm(float x) { return 1.0f / (1.0f + expf(-x)); }

// CDNA5 async DMA: copy 16B from global to LDS, tracked by ASYNCcnt.
__device__ __forceinline__ void async_load16(void* lds, const void* g) {
    uint32_t l  = (uint32_t)(uintptr_t)lds;       // flat addr[31:0] == LDS offset
    uint64_t ga = (uint64_t)(uintptr_t)g;
    asm volatile("global_load_async_to_lds_b128 %0, %1, off"
                 :: "v"(l), "v"(ga) : "memory");
}
__device__ __forceinline__ void wait_async0() {
    asm volatile("s_wait_asynccnt 0x0" ::: "memory");
}

// ---------------------------------------------------------------------------
// Kernel 1: stability / emergence / decay.  One wave32 per (l, s).
// ---------------------------------------------------------------------------
__global__ void stats_kernel(const float* __restrict__ feat, float* __restrict__ out) {
    int wave = (blockIdx.x * blockDim.x + threadIdx.x) >> 5;
    int lane = threadIdx.x & 31;
    if (wave >= STAB_N) return;
    int l = wave / 512, s = wave % 512;
    float stab = 0.f, em = 0.f;
    for (int b = 0; b < 2; ++b) {
        const float* pprev = feat + ((size_t)((b * 8 + l) * 512 + s) << 8);
        const float* pnxt  = feat + ((size_t)((b * 8 + l + 1) * 512 + s) << 8);
        float dot = 0.f, na = 0.f, nb = 0.f, ap = 0.f, an = 0.f;
        for (int d = lane; d < 256; d += 32) {
            float pv = pprev[d], nv = pnxt[d];
            dot += pv * nv; na += pv * pv; nb += nv * nv;
            ap += (pv > 0.1f) ? 1.0f : 0.0f;
            an += (nv > 0.1f) ? 1.0f : 0.0f;
        }
        for (int m = 16; m; m >>= 1) {           // wave32 butterfly
            dot += __shfl_xor(dot, m, 32);
            na  += __shfl_xor(na,  m, 32);
            nb  += __shfl_xor(nb,  m, 32);
            ap  += __shfl_xor(ap,  m, 32);
            an  += __shfl_xor(an,  m, 32);
        }
        float denom = fmaxf(sqrtf(na), 1e-8f) * fmaxf(sqrtf(nb), 1e-8f);
        stab += dot / denom;
        em   += (an - ap) * (1.0f / 256.0f);
    }
    if (lane == 0) {
        stab *= 0.5f; em *= 0.5f;
        out[wave]              = stab;   // stability
        out[STAB_N + wave]     = em;     // emergence
        out[2 * STAB_N + wave] = -em;    // decay == -emergence exactly
    }
}

// ---------------------------------------------------------------------------
// Kernel 2: A[slice, o] = src[slice,:] . W1[o, :256] + b1[o]   (64 blocks x 16 slices)
// ---------------------------------------------------------------------------
__global__ void a_kernel(const float* __restrict__ src, const float* __restrict__ W1,
                         const float* __restrict__ b1, float* __restrict__ A_ws) {
    __shared__ float sl[16 * 256];
    int tid = threadIdx.x;           // 256
    int s0  = blockIdx.x * 16;
    for (int idx = tid; idx < 16 * 256; idx += 256)
        sl[idx] = src[((size_t)s0 << 8) + idx];
    __syncthreads();
    float acc[16];
    float bo = b1[tid];
#pragma unroll
    for (int i = 0; i < 16; ++i) acc[i] = bo;
    const float* wr = W1 + (size_t)tid * 512;
    for (int k = 0; k < 256; ++k) {
        float w = wr[k];
#pragma unroll
        for (int i = 0; i < 16; ++i) acc[i] += w * sl[i * 256 + k];
    }
#pragma unroll
    for (int i = 0; i < 16; ++i) A_ws[((size_t)(s0 + i) << 8) + tid] = acc[i];
}

// ---------------------------------------------------------------------------
// Kernel 3: wb[o] = sum_j W1[o, 256+j]
// ---------------------------------------------------------------------------
__global__ void wb_kernel(const float* __restrict__ W1, float* __restrict__ wb_ws) {
    int o = threadIdx.x;
    const float* row = W1 + (size_t)o * 512 + 256;
    float s = 0.f;
    for (int j = 0; j < 256; ++j) s += row[j];
    wb_ws[o] = s;
}

// ---------------------------------------------------------------------------
// Kernel 4: causal GEMM with v_wmma_f32_16x16x4_f32 + async-to-LDS pipelining.
// Grid: 256 blocks = (4 i-tiles x 4 p-tiles) x 16 slice-groups; 512 threads = 16 waves.
// W2 tile staged row-major [p_local][k] stride 260 (async-DMA friendly,
// conflict-free: 260 % 64 == 4).  A_s/t_s double-buffered: DMA for slice sl+1
// overlaps the 64-WMMA K-loop of slice sl.
// ---------------------------------------------------------------------------
__global__ __launch_bounds__(512)
void causal_kernel(const float* __restrict__ tgt, const float* __restrict__ W2,
                   const float* __restrict__ b2, const float* __restrict__ A_ws,
                   const float* __restrict__ wb_ws, float* __restrict__ partials) {
    __shared__ float w2r[64 * 260];   // [p_local][k], stride 260
    __shared__ float A_s[2][256];
    __shared__ float t_s[2][64];
    __shared__ float wb_s[256];
    __shared__ float b2_s[64];

    int tid  = threadIdx.x;
    int tile = blockIdx.x & 15;
    int grp  = blockIdx.x >> 4;       // slice group 0..15
    int i0   = (tile >> 2) * 64;
    int p0   = (tile & 3) * 64;

    // Async-stage W2 tile: 64 rows x 64 16B-chunks.
    for (int idx = tid; idx < 4096; idx += 512) {
        int pl = idx >> 6, kc = idx & 63;
        async_load16(&w2r[pl * 260 + kc * 4],
                     W2 + ((size_t)(p0 + pl) << 8) + kc * 4);
    }
    // Async-stage slice 0 (buffer 0).
    {
        int slice0 = grp * 64;
        if (tid < 64)
            async_load16(&A_s[0][tid * 4], A_ws + ((size_t)slice0 << 8) + tid * 4);
        else if (tid < 80)
            async_load16(&t_s[0][(tid - 64) * 4],
                         tgt + ((size_t)slice0 << 8) + i0 + (tid - 64) * 4);
    }
    if (tid < 256) wb_s[tid] = wb_ws[tid];
    if (tid < 64)  b2_s[tid] = b2[p0 + tid];
    wait_async0();
    __syncthreads();

    int w    = tid >> 5;              // wave 0..15
    int lane = tid & 31;
    int wi   = w >> 2, wp = w & 3;    // 16-row / 16-col sub-tile within 64x64
    int half = lane >> 4;             // K sub-phase per A/B 32-bit layout
    int n    = lane & 15;
    float b2v = b2_s[wp * 16 + n];
    const float* brow = &w2r[(wp * 16 + n) * 260];  // this lane's W2 row (B[k][n])
    v8f acc = {};

    for (int sl = 0; sl < 64; ++sl) {
        int cur = sl & 1;
        // Prefetch next slice into the other buffer via async DMA.
        if (sl + 1 < 64) {
            int nsl = grp * 64 + sl + 1;
            if (tid < 64)
                async_load16(&A_s[cur ^ 1][tid * 4], A_ws + ((size_t)nsl << 8) + tid * 4);
            else if (tid < 80)
                async_load16(&t_s[cur ^ 1][(tid - 64) * 4],
                             tgt + ((size_t)nsl << 8) + i0 + (tid - 64) * 4);
        }

        float t_m = t_s[cur][wi * 16 + n];  // A-matrix M index == lane&15
        v8f c = {};
#pragma unroll 4
        for (int kb = 0; kb < 256; kb += 4) {
            int k0 = kb + 2 * half;
            v2f a, bm;
            a.x  = fmaxf(A_s[cur][k0]     + t_m * wb_s[k0],     0.0f); // h = relu(A+t*wb)
            a.y  = fmaxf(A_s[cur][k0 + 1] + t_m * wb_s[k0 + 1], 0.0f);
            bm.x = brow[k0];                                           // B[k][n] = W2[p,k]
            bm.y = brow[k0 + 1];
            c = __builtin_amdgcn_wmma_f32_16x16x4_f32(
                    false, a, false, bm, (short)0, c, false, false);
        }
#pragma unroll
        for (int r = 0; r < 8; ++r)
            acc[r] += sigm(c[r] + b2v);

        wait_async0();
        __syncthreads();
    }

    float* pb = partials + (size_t)grp * 65536;
#pragma unroll
    for (int r = 0; r < 8; ++r) {
        int i = i0 + wi * 16 + half * 8 + r;    // C/D layout: VGPR r -> M = r + 8*half
        int p = p0 + wp * 16 + n;
        pb[i * 256 + p] = acc[r];
    }
}

// ---------------------------------------------------------------------------
// Kernel 5: reduce 16 slice-group partials -> causal_matrix (/1024)
// ---------------------------------------------------------------------------
__global__ void reduce_kernel(const float* __restrict__ partials, float* __restrict__ out) {
    int idx = blockIdx.x * 256 + threadIdx.x;   // 0..65535
    float s = 0.f;
#pragma unroll
    for (int g = 0; g < 16; ++g) s += partials[(size_t)g * 65536 + idx];
    out[CAUSAL_OFF + idx] = s * (1.0f / 1024.0f);
}

// ---------------------------------------------------------------------------
// Kernel 6: GRU. 64 blocks x 1024 threads; block owns 16 batch rows, h lives in
// LDS across all 8 timesteps. Thread groups: 0=r(concat K), 1=z(concat K),
// 2=gx_n (x only), 3=gh_n (h only)  — n-gate mixes gx/gh differently.
// ---------------------------------------------------------------------------
__global__ __launch_bounds__(1024)
void gru_kernel(const float* __restrict__ feat, const float* __restrict__ W_ih,
                const float* __restrict__ W_hh, const float* __restrict__ b_ih,
                const float* __restrict__ b_hh, float* __restrict__ evo) {
    __shared__ float xh[512 * 17];    // [k][m]: k<256 = x_t, k>=256 = h   (stride 17)
    __shared__ float Gm[1024 * 17];   // [gate-slot][m]

    int tid = threadIdx.x;
    int n0  = blockIdx.x * 16;
    int grp = tid >> 8, g = tid & 255;

    // h = 0
    for (int idx = tid; idx < 4096; idx += 1024) {
        int k = 256 + (idx >> 4), m = idx & 15;
        xh[k * 17 + m] = 0.0f;
    }

    float b_r = 0.f, b_z = 0.f, bnx = 0.f, bnh = 0.f;
    if (tid < 256) {
        b_r = b_ih[g]       + b_hh[g];
        b_z = b_ih[256 + g] + b_hh[256 + g];
        bnx = b_ih[512 + g];
        bnh = b_hh[512 + g];
    }

    for (int t = 0; t < 8; ++t) {
        // stage x_t (transposed into [k][m])
        for (int idx = tid; idx < 4096; idx += 1024) {
            int m = idx >> 8, k = idx & 255;
            int nn = n0 + m;
            int b = nn >> 9, s = nn & 511;
            xh[k * 17 + m] = feat[((size_t)((b * 8 + t) * 512 + s) << 8) + k];
        }
        __syncthreads();

        float acc[16];
#pragma unroll
        for (int m = 0; m < 16; ++m) acc[m] = 0.0f;

        if (grp <= 1) {                    // r / z gates: K = [x | h] (512)
            const float* wa = W_ih + (size_t)(grp * 256 + g) * 256;
            const float* wh = W_hh + (size_t)(grp * 256 + g) * 256;
            for (int k = 0; k < 256; ++k) {
                float w = wa[k];
#pragma unroll
                for (int m = 0; m < 16; ++m) acc[m] += w * xh[k * 17 + m];
            }
            for (int k = 0; k < 256; ++k) {
                float w = wh[k];
#pragma unroll
                for (int m = 0; m < 16; ++m) acc[m] += w * xh[(256 + k) * 17 + m];
            }
        } else if (grp == 2) {             // gx_n
            const float* wa = W_ih + (size_t)(512 + g) * 256;
            for (int k = 0; k < 256; ++k) {
                float w = wa[k];
#pragma unroll
                for (int m = 0; m < 16; ++m) acc[m] += w * xh[k * 17 + m];
            }
        } else {                           // gh_n
            const float* wh = W_hh + (size_t)(512 + g) * 256;
            for (int k = 0; k < 256; ++k) {
                float w = wh[k];
#pragma unroll
                for (int m = 0; m < 16; ++m) acc[m] += w * xh[(256 + k) * 17 + m];
            }
        }
#pragma unroll
        for (int m = 0; m < 16; ++m) Gm[tid * 17 + m] = acc[m];
        __syncthreads();

        if (tid < 256) {
#pragma unroll 4
            for (int m = 0; m < 16; ++m) {
                float r  = sigm(Gm[g * 17 + m] + b_r);
                float z  = sigm(Gm[(256 + g) * 17 + m] + b_z);
                float nn = tanhf(Gm[(512 + g) * 17 + m] + bnx +
                                 r * (Gm[(768 + g) * 17 + m] + bnh));
                float hold = xh[(256 + g) * 17 + m];
                float hnew = (1.0f - z) * nn + z * hold;
                xh[(256 + g) * 17 + m] = hnew;
                evo[((size_t)((n0 + m) * 8 + t) << 8) + g] = hnew;
            }
        }
        // next-iteration barrier (after x staging) orders h-update vs GEMM reads
    }
}

// ---------------------------------------------------------------------------
extern "C" void kernel_launch(void* const* d_in, const int* in_sizes, int n_in,
                              void* d_out, int out_size, void* d_ws, size_t ws_size,
                              hipStream_t stream) {
    const float* feat = (const float*)d_in[0];
    const float* src  = (const float*)d_in[1];
    const float* tgt  = (const float*)d_in[2];
    const float* W_ih = (const float*)d_in[3];
    const float* W_hh = (const float*)d_in[4];
    const float* b_ih = (const float*)d_in[5];
    const float* b_hh = (const float*)d_in[6];
    const float* W1   = (const float*)d_in[7];
    const float* b1   = (const float*)d_in[8];
    const float* W2   = (const float*)d_in[9];
    const float* b2   = (const float*)d_in[10];
    float* out = (float*)d_out;

    float* A_ws     = (float*)d_ws;          // 1024*256 floats
    float* wb_ws    = A_ws + 262144;         // 256 floats
    float* partials = wb_ws + 256;           // 16*65536 floats

    stats_kernel <<<448, 256, 0, stream>>>(feat, out);
    a_kernel     <<<64,  256, 0, stream>>>(src, W1, b1, A_ws);
    wb_kernel    <<<1,   256, 0, stream>>>(W1, wb_ws);
    causal_kernel<<<256, 512, 0, stream>>>(tgt, W2, b2, A_ws, wb_ws, partials);
    reduce_kernel<<<256, 256, 0, stream>>>(partials, out);
    gru_kernel   <<<64, 1024, 0, stream>>>(feat, W_ih, W_hh, b_ih, b_hh, out + EVO_OFF);
}